// InductiveCriticNetwork_1915555414398
// MI455X (gfx1250) — compile-verified
//
#include <hip/hip_runtime.h>
#include <hip/hip_bf16.h>

#define NNODES 50000
#define NEDGES 800000
#define BSZ    7680
#define TOK    (BSZ * 9)     // 3 servers + 6 users valid per graph

typedef float v2f __attribute__((ext_vector_type(2)));
typedef float v8f __attribute__((ext_vector_type(8)));

// ---------------- elementwise helpers ----------------
__global__ void fill_kernel(float* p, float v, int n) {
    int i = blockIdx.x * blockDim.x + threadIdx.x;
    if (i < n) p[i] = v;
}

__global__ void deg_edges_kernel(const int* __restrict__ dst, float* deg, int E) {
    int e = blockIdx.x * blockDim.x + threadIdx.x;
    if (e < E) atomicAdd(&deg[dst[e]], 1.0f);
}

__global__ void rsqrt_kernel(float* p, int n) {
    int i = blockIdx.x * blockDim.x + threadIdx.x;
    if (i < n) p[i] = rsqrtf(p[i]);
}

__global__ void relu_kernel(float* p, int n) {
    int i = blockIdx.x * blockDim.x + threadIdx.x;
    if (i < n) p[i] = fmaxf(p[i], 0.0f);
}

__global__ void axpy_kernel(float* g, const float* gd, int n) {
    int i = blockIdx.x * blockDim.x + threadIdx.x;
    if (i < n) g[i] += gd[i];
}

// g0 = global_vec @ Wgs + bgs   ([B,3] @ [3,64])
__global__ void g0_kernel(const float* __restrict__ gv, const float* __restrict__ Wgs,
                          const float* __restrict__ bgs, float* g, int Btot) {
    int i = blockIdx.x * blockDim.x + threadIdx.x;
    if (i >= Btot * 64) return;
    int b = i >> 6, j = i & 63;
    float s = bgs[j];
    #pragma unroll
    for (int k = 0; k < 3; ++k) s += gv[b * 3 + k] * Wgs[k * 64 + j];
    g[i] = s;
}

// token t of graph b: j<3 -> servers[b*3+j], else users[b*6+(j-3)]
__global__ void tokens_kernel(const int* __restrict__ servers, const int* __restrict__ users,
                              int* tok, int Btot) {
    int i = blockIdx.x * blockDim.x + threadIdx.x;
    if (i >= Btot * 9) return;
    int b = i / 9, j = i % 9;
    tok[i] = (j < 3) ? servers[b * 3 + j] : users[b * 6 + (j - 3)];
}

// cat[b, 0:H) = 0 (reduction target), cat[b, H:H+64) = g[b]
__global__ void prep_cat_kernel(const float* __restrict__ g, float* cat, int H, int Btot) {
    int stride = H + 64;
    int i = blockIdx.x * blockDim.x + threadIdx.x;
    if (i >= Btot * stride) return;
    int b = i / stride, c = i % stride;
    cat[i] = (c < H) ? 0.0f : g[b * 64 + (c - H)];
}

// h_agg = bias + hlin * dinv^2   (self-loop term + bias init)
__global__ void gcn_init_kernel(const float* __restrict__ hlin, const float* __restrict__ dinv,
                                const float* __restrict__ bias, float* out, int n, int D) {
    int i = blockIdx.x * blockDim.x + threadIdx.x;
    if (i >= n * D) return;
    int node = i / D, f = i % D;
    float dv = dinv[node];
    out[i] = bias[f] + hlin[i] * dv * dv;
}

// one wave per edge: out[dst] += hlin[src] * dinv[src]*dinv[dst]
template <int D>
__global__ void gcn_edges_kernel(const int* __restrict__ esrc, const int* __restrict__ edst,
                                 const float* __restrict__ hlin, const float* __restrict__ dinv,
                                 float* out, int E) {
    int wave = threadIdx.x >> 5, lane = threadIdx.x & 31;
    int e = blockIdx.x * (blockDim.x >> 5) + wave;
    if (e >= E) return;
    int s = esrc[e], d = edst[e];
    float nrm = dinv[s] * dinv[d];
    const float* hp = hlin + (long)s * D;
    float* op = out + (long)d * D;
    #pragma unroll
    for (int r = 0; r < D / 32; ++r)
        atomicAdd(&op[lane + r * 32], hp[lane + r * 32] * nrm);
}

// ---------------- WMMA fp32 GEMM: C = act(gather(A) @ W + bias) ----------------
// Wave computes a 32 x (16*NT) tile: two 16-row A fragments share every B fragment,
// so each B load feeds 2 WMMAs (halves VMEM per matrix op vs a 16-row tile).
// A-frag (16x4 f32): lane<16 holds {A[m][k0],A[m][k0+1]}, lane>=16 holds {A[m][k0+2],A[m][k0+3]}
// C-frag: reg r, lanes0-15 -> M=r, lanes16-31 -> M=r+8; N = lane&15.
// All callers have M % 16 == 0, so the second sub-tile is uniformly valid or invalid.
template <int NT, int ACT>
__global__ void gemm_wmma_kernel(const float* __restrict__ A, const int* __restrict__ idx,
                                 const float* __restrict__ W, const float* __restrict__ bias,
                                 float* __restrict__ C, int M, int K, int N) {
    int wave = threadIdx.x >> 5, lane = threadIdx.x & 31;
    int tile = blockIdx.x * (blockDim.x >> 5) + wave;  // 32-row tile index
    int m0 = tile * 32;
    if (m0 >= M) return;
    bool has1 = (m0 + 16) < M;                         // uniform (M % 16 == 0)
    int n0 = blockIdx.y * 16 * NT;
    int l15 = lane & 15;
    int koff = (lane >> 4) << 1;

    int mrow0 = m0 + l15;
    int mrow1 = has1 ? (m0 + 16 + l15) : mrow0;
    int row0 = idx ? idx[mrow0] : mrow0;
    int row1 = idx ? idx[mrow1] : mrow1;
    const float* ap0 = A + (long)row0 * K + koff;
    const float* ap1 = A + (long)row1 * K + koff;

    int ncol[NT];
    v8f acc0[NT], acc1[NT];
    #pragma unroll
    for (int t = 0; t < NT; ++t) {
        ncol[t] = n0 + t * 16 + l15;
        float bv = bias ? bias[ncol[t]] : 0.0f;
        acc0[t] = (v8f){bv, bv, bv, bv, bv, bv, bv, bv};
        acc1[t] = acc0[t];
    }
    for (int k0 = 0; k0 < K; k0 += 4) {
        v2f a0 = *(const v2f*)(ap0 + k0);
        v2f a1 = *(const v2f*)(ap1 + k0);
        const float* wrow = W + (long)(k0 + koff) * N;
        #pragma unroll
        for (int t = 0; t < NT; ++t) {
            v2f b;
            b.x = wrow[ncol[t]];
            b.y = wrow[N + ncol[t]];
            acc0[t] = __builtin_amdgcn_wmma_f32_16x16x4_f32(
                false, a0, false, b, (short)0, acc0[t], false, false);
            acc1[t] = __builtin_amdgcn_wmma_f32_16x16x4_f32(
                false, a1, false, b, (short)0, acc1[t], false, false);
        }
    }
    int mb0 = m0 + ((lane >> 4) << 3);
    #pragma unroll
    for (int t = 0; t < NT; ++t) {
        #pragma unroll
        for (int r = 0; r < 8; ++r) {
            float v = acc0[t][r];
            if (ACT == 1) v = fmaxf(v, 0.0f);
            if (ACT == 2) v = tanhf(v);
            C[(long)(mb0 + r) * N + ncol[t]] = v;
        }
    }
    if (has1) {
        int mb1 = mb0 + 16;
        #pragma unroll
        for (int t = 0; t < NT; ++t) {
            #pragma unroll
            for (int r = 0; r < 8; ++r) {
                float v = acc1[t][r];
                if (ACT == 1) v = fmaxf(v, 0.0f);
                if (ACT == 2) v = tanhf(v);
                C[(long)(mb1 + r) * N + ncol[t]] = v;
            }
        }
    }
}

// ---------------- fused softmax(attL) * featL -> atomic sum into out[b] ----------------
template <int H>
__global__ void attn_reduce_kernel(const float* __restrict__ attL, const float* __restrict__ featL,
                                   float* outcat, int catStride, int T) {
    int wave = threadIdx.x >> 5, lane = threadIdx.x & 31;
    int t = blockIdx.x * (blockDim.x >> 5) + wave;
    if (t >= T) return;
    constexpr int R = H / 32;
    const float* ap = attL + (long)t * H;
    float a[R];
    float mx = -3.0e38f;
    #pragma unroll
    for (int r = 0; r < R; ++r) { a[r] = ap[lane + r * 32]; mx = fmaxf(mx, a[r]); }
    #pragma unroll
    for (int m = 16; m >= 1; m >>= 1) mx = fmaxf(mx, __shfl_xor(mx, m, 32));
    float s = 0.0f;
    #pragma unroll
    for (int r = 0; r < R; ++r) { a[r] = expf(a[r] - mx); s += a[r]; }
    #pragma unroll
    for (int m = 16; m >= 1; m >>= 1) s += __shfl_xor(s, m, 32);
    float inv = 1.0f / s;
    const float* fp = featL + (long)t * H;
    float* op = outcat + (long)(t / 9) * catStride;
    #pragma unroll
    for (int r = 0; r < R; ++r)
        atomicAdd(&op[lane + r * 32], a[r] * inv * fp[lane + r * 32]);
}

// ---------------- final: group-mean(3) -> relu(@Wo1+bo1) -> @Wo2+bo2 ----------------
__global__ void final_kernel(const float* __restrict__ g, const float* __restrict__ Wo1,
                             const float* __restrict__ bo1, const float* __restrict__ Wo2,
                             const float* __restrict__ bo2, float* out) {
    int grp = blockIdx.x, lane = threadIdx.x;  // 32 threads
    __shared__ float gm[64];
    for (int j = lane; j < 64; j += 32) {
        float s = g[(grp * 3 + 0) * 64 + j] + g[(grp * 3 + 1) * 64 + j] + g[(grp * 3 + 2) * 64 + j];
        gm[j] = s * (1.0f / 3.0f);
    }
    __syncthreads();
    float t = bo1[lane];
    #pragma unroll
    for (int k = 0; k < 64; ++k) t += gm[k] * Wo1[k * 32 + lane];
    t = fmaxf(t, 0.0f);
    float p = t * Wo2[lane];
    #pragma unroll
    for (int m = 16; m >= 1; m >>= 1) p += __shfl_xor(p, m, 32);
    if (lane == 0) out[grp] = p + bo2[0];
}

static inline int cdiv(int a, int b) { return (a + b - 1) / b; }

extern "C" void kernel_launch(void* const* d_in, const int* in_sizes, int n_in,
                              void* d_out, int out_size, void* d_ws, size_t ws_size,
                              hipStream_t stream) {
    const float* x    = (const float*)d_in[0];
    const int*   ei   = (const int*)d_in[1];
    const int*   esrc = ei;
    const int*   edst = ei + NEDGES;
    const float* gv   = (const float*)d_in[2];
    const int* servers = (const int*)d_in[3];
    const int* users   = (const int*)d_in[5];
    const float* Wc1 = (const float*)d_in[9],  *bc1 = (const float*)d_in[10];
    const float* Wc2 = (const float*)d_in[11], *bc2 = (const float*)d_in[12];
    const float* Wgs = (const float*)d_in[13], *bgs = (const float*)d_in[14];
    const float* Wa0 = (const float*)d_in[15], *ba0 = (const float*)d_in[16];
    const float* Wf0 = (const float*)d_in[17], *bf0 = (const float*)d_in[18];
    const float* Wg0 = (const float*)d_in[19], *bg0 = (const float*)d_in[20];
    const float* Wa1 = (const float*)d_in[21], *ba1 = (const float*)d_in[22];
    const float* Wf1 = (const float*)d_in[23], *bf1 = (const float*)d_in[24];
    const float* Wg1 = (const float*)d_in[25], *bg1 = (const float*)d_in[26];
    const float* Wa2 = (const float*)d_in[27], *ba2 = (const float*)d_in[28];
    const float* Wf2 = (const float*)d_in[29], *bf2 = (const float*)d_in[30];
    const float* Wg2 = (const float*)d_in[31], *bg2 = (const float*)d_in[32];
    const float* Wo1 = (const float*)d_in[33], *bo1 = (const float*)d_in[34];
    const float* Wo2 = (const float*)d_in[35], *bo2 = (const float*)d_in[36];
    (void)in_sizes; (void)n_in; (void)out_size; (void)ws_size;

    // workspace layout (floats), 256-element aligned
    float* ws = (float*)d_ws;
    size_t off = 0;
    auto alloc = [&](size_t n) { float* p = ws + off; off += (n + 255) & ~(size_t)255; return p; };
    float* dinv = alloc(NNODES);
    int*   tok  = (int*)alloc(TOK);
    float* g    = alloc((size_t)BSZ * 64);
    float* gd   = alloc((size_t)BSZ * 64);
    float* cat  = alloc((size_t)BSZ * 320);
    float* bufA = alloc((size_t)TOK * 256);         // att logits / h2lin (reused)
    float* bufB = alloc((size_t)TOK * 256);         // feat / hlin (reused)
    float* h1   = alloc((size_t)NNODES * 256);
    float* h2   = alloc((size_t)NNODES * 64);

    // 32-row GEMM tiles: blocks of 4 waves
    int tilesTok = cdiv(cdiv(TOK, 32), 4);          // 540
    int tilesNod = cdiv(cdiv(NNODES, 32), 4);       // 391 (last tile half-valid)
    int tilesB   = cdiv(cdiv(BSZ, 32), 4);          // 60

    // --- degree -> dinv (self-loop => init 1.0) ---
    fill_kernel<<<cdiv(NNODES, 256), 256, 0, stream>>>(dinv, 1.0f, NNODES);
    deg_edges_kernel<<<cdiv(NEDGES, 256), 256, 0, stream>>>(edst, dinv, NEDGES);
    rsqrt_kernel<<<cdiv(NNODES, 256), 256, 0, stream>>>(dinv, NNODES);

    g0_kernel<<<cdiv(BSZ * 64, 256), 256, 0, stream>>>(gv, Wgs, bgs, g, BSZ);
    tokens_kernel<<<cdiv(TOK, 256), 256, 0, stream>>>(servers, users, tok, BSZ);

    // --- attention layer 0 on x (d=128, H=256) ---
    gemm_wmma_kernel<4, 0><<<dim3(tilesTok, 4), 128, 0, stream>>>(x, tok, Wa0, ba0, bufA, TOK, 128, 256);
    gemm_wmma_kernel<4, 0><<<dim3(tilesTok, 4), 128, 0, stream>>>(x, tok, Wf0, bf0, bufB, TOK, 128, 256);
    prep_cat_kernel<<<cdiv(BSZ * 320, 256), 256, 0, stream>>>(g, cat, 256, BSZ);
    attn_reduce_kernel<256><<<cdiv(TOK, 8), 256, 0, stream>>>(bufA, bufB, cat, 320, TOK);
    gemm_wmma_kernel<4, 2><<<dim3(tilesB, 1), 128, 0, stream>>>(cat, nullptr, Wg0, bg0, gd, BSZ, 320, 64);
    axpy_kernel<<<cdiv(BSZ * 64, 256), 256, 0, stream>>>(g, gd, BSZ * 64);

    // --- GCN layer 1: h1 = relu(agg(x @ Wc1) + bc1) ---
    gemm_wmma_kernel<4, 0><<<dim3(tilesNod, 4), 128, 0, stream>>>(x, nullptr, Wc1, nullptr, bufB, NNODES, 128, 256);
    gcn_init_kernel<<<cdiv(NNODES * 256, 256), 256, 0, stream>>>(bufB, dinv, bc1, h1, NNODES, 256);
    gcn_edges_kernel<256><<<cdiv(NEDGES, 8), 256, 0, stream>>>(esrc, edst, bufB, dinv, h1, NEDGES);
    relu_kernel<<<cdiv(NNODES * 256, 256), 256, 0, stream>>>(h1, NNODES * 256);

    // --- attention layer 1 on h1 (d=256, H=256) ---
    gemm_wmma_kernel<4, 0><<<dim3(tilesTok, 4), 128, 0, stream>>>(h1, tok, Wa1, ba1, bufA, TOK, 256, 256);
    gemm_wmma_kernel<4, 0><<<dim3(tilesTok, 4), 128, 0, stream>>>(h1, tok, Wf1, bf1, bufB, TOK, 256, 256);
    prep_cat_kernel<<<cdiv(BSZ * 320, 256), 256, 0, stream>>>(g, cat, 256, BSZ);
    attn_reduce_kernel<256><<<cdiv(TOK, 8), 256, 0, stream>>>(bufA, bufB, cat, 320, TOK);
    gemm_wmma_kernel<4, 2><<<dim3(tilesB, 1), 128, 0, stream>>>(cat, nullptr, Wg1, bg1, gd, BSZ, 320, 64);
    axpy_kernel<<<cdiv(BSZ * 64, 256), 256, 0, stream>>>(g, gd, BSZ * 64);

    // --- GCN layer 2: h2 = relu(agg(h1 @ Wc2) + bc2) ---
    gemm_wmma_kernel<4, 0><<<dim3(tilesNod, 1), 128, 0, stream>>>(h1, nullptr, Wc2, nullptr, bufA, NNODES, 256, 64);
    gcn_init_kernel<<<cdiv(NNODES * 64, 256), 256, 0, stream>>>(bufA, dinv, bc2, h2, NNODES, 64);
    gcn_edges_kernel<64><<<cdiv(NEDGES, 8), 256, 0, stream>>>(esrc, edst, bufA, dinv, h2, NEDGES);
    relu_kernel<<<cdiv(NNODES * 64, 256), 256, 0, stream>>>(h2, NNODES * 64);

    // --- attention layer 2 on h2 (d=64, H=64) ---
    gemm_wmma_kernel<4, 0><<<dim3(tilesTok, 1), 128, 0, stream>>>(h2, tok, Wa2, ba2, bufA, TOK, 64, 64);
    gemm_wmma_kernel<4, 0><<<dim3(tilesTok, 1), 128, 0, stream>>>(h2, tok, Wf2, bf2, bufB, TOK, 64, 64);
    prep_cat_kernel<<<cdiv(BSZ * 128, 256), 256, 0, stream>>>(g, cat, 64, BSZ);
    attn_reduce_kernel<64><<<cdiv(TOK, 8), 256, 0, stream>>>(bufA, bufB, cat, 128, TOK);
    gemm_wmma_kernel<4, 2><<<dim3(tilesB, 1), 128, 0, stream>>>(cat, nullptr, Wg2, bg2, gd, BSZ, 128, 64);
    axpy_kernel<<<cdiv(BSZ * 64, 256), 256, 0, stream>>>(g, gd, BSZ * 64);

    // --- final head: group mean of 3 -> relu(@Wo1) -> @Wo2 ---
    final_kernel<<<BSZ / 3, 32, 0, stream>>>(g, Wo1, bo1, Wo2, bo2, (float*)d_out);
}